// MainNet_55877524521400
// MI455X (gfx1250) — compile-verified
//
#include <hip/hip_runtime.h>
#include <hip/hip_bf16.h>

#define BN      8
#define FM      28
#define IMGSZ   448
#define STRIDEPX 16
#define NPROP   7
#define TOT_WIN 1595
#define NEG_INF (-__builtin_inff())

typedef float v2f __attribute__((ext_vector_type(2)));
typedef float v8f __attribute__((ext_vector_type(8)));

// ---------------------------------------------------------------- channel sum
__global__ void k_sumch(const float* __restrict__ fm, float* __restrict__ s) {
    int t = blockIdx.x * blockDim.x + threadIdx.x;
    if (t >= BN * FM * FM) return;
    int b = t / (FM * FM);
    int pos = t - b * (FM * FM);
    const float* p = fm + (size_t)b * 384 * FM * FM + pos;
    float acc = 0.f;
    for (int c = 0; c < 384; ++c) acc += p[(size_t)c * FM * FM];
    s[t] = acc;
}

// ---------------------------------------------------------------- window avg scores
__global__ void k_scores(const float* __restrict__ s, float* __restrict__ all) {
    int t = blockIdx.x * blockDim.x + threadIdx.x;
    if (t >= BN * TOT_WIN) return;
    int b = t / TOT_WIN;
    int w = t - b * TOT_WIN;
    int off, n, r;
    if (w < 625)       { off = 0;    n = 25; r = 4; }
    else if (w < 1154) { off = 625;  n = 23; r = 6; }
    else               { off = 1154; n = 21; r = 8; }
    int local = w - off;
    int ii = local / n, jj = local - ii * n;
    const float* sp = s + b * FM * FM + ii * FM + jj;
    float acc = 0.f;
    for (int di = 0; di < r; ++di)
        for (int dj = 0; dj < r; ++dj)
            acc += sp[di * FM + dj];
    all[t] = acc / (float)(r * r);
}

// ---------------------------------------------------------------- greedy NMS per (batch, group)
__global__ void k_nms(const float* __restrict__ all,
                      float* __restrict__ out_idx, float* __restrict__ out_sc,
                      int* __restrict__ sel) {
    __shared__ float sc[640];
    __shared__ float rv[256];
    __shared__ int   ri[256];
    __shared__ float selbox[4];

    int task = blockIdx.x;
    int b = task / 3, g = task - 3 * (task / 3);
    int off, ng, n, r, nsel, col0;
    if (g == 0)      { off = 0;    ng = 625; n = 25; r = 4; nsel = 2; col0 = 0; }
    else if (g == 1) { off = 625;  ng = 529; n = 23; r = 6; nsel = 3; col0 = 2; }
    else             { off = 1154; ng = 441; n = 21; r = 8; nsel = 2; col0 = 5; }

    int tid = threadIdx.x;
    for (int i = tid; i < ng; i += 256) sc[i] = all[b * TOT_WIN + off + i];
    __syncthreads();

    float side = (float)(r * STRIDEPX);          // box side (x1 - x0)
    float area = (side + 1.f) * (side + 1.f);    // inclusive area, same for all boxes in group

    for (int k = 0; k < nsel; ++k) {
        // --- argmax with first-index tie-break (matches jnp.argmax)
        float bv = NEG_INF; int bi = ng;
        for (int i = tid; i < ng; i += 256) {
            float v = sc[i];
            if (v > bv || (v == bv && i < bi)) { bv = v; bi = i; }
        }
        rv[tid] = bv; ri[tid] = bi;
        __syncthreads();
        for (int st = 128; st > 0; st >>= 1) {
            if (tid < st) {
                float ov = rv[tid + st]; int oi = ri[tid + st];
                if (ov > rv[tid] || (ov == rv[tid] && oi < ri[tid])) { rv[tid] = ov; ri[tid] = oi; }
            }
            __syncthreads();
        }
        if (tid == 0) {
            int i = ri[0];
            int ii = i / n, jj = i - ii * n;
            selbox[0] = (float)(ii * STRIDEPX);
            selbox[1] = (float)(jj * STRIDEPX);
            selbox[2] = selbox[0] + side;
            selbox[3] = selbox[1] + side;
            out_idx[b * NPROP + col0 + k] = (float)(i + off);
            out_sc [b * NPROP + col0 + k] = sc[i];          // original (pre-suppression) score
            sel    [b * NPROP + col0 + k] = i + off;
        }
        __syncthreads();
        float bx0 = selbox[0], by0 = selbox[1], bx1 = selbox[2], by1 = selbox[3];
        for (int i = tid; i < ng; i += 256) {
            int ii = i / n, jj = i - ii * n;
            float x0 = (float)(ii * STRIDEPX), y0 = (float)(jj * STRIDEPX);
            float x1 = x0 + side, y1 = y0 + side;
            float xx0 = fmaxf(x0, bx0), yy0 = fmaxf(y0, by0);
            float xx1 = fminf(x1, bx1), yy1 = fminf(y1, by1);
            float w_ = xx1 - xx0 + 1.f, h_ = yy1 - yy0 + 1.f;
            float inter = (w_ < 0.f || h_ < 0.f) ? 0.f : w_ * h_;
            float iou = inter / (area + area - inter);
            if (iou > 0.25f) sc[i] = NEG_INF;
        }
        __syncthreads();
    }
}

// ---------------------------------------------------------------- crop + bilinear resize
__global__ void k_crop(const float* __restrict__ x, const int* __restrict__ coord,
                       const int* __restrict__ sel, float* __restrict__ out) {
    int t = blockIdx.x * blockDim.x + threadIdx.x;   // total = 8*7*3*448*448 < 2^31
    const int total = BN * NPROP * 3 * IMGSZ * IMGSZ;
    if (t >= total) return;
    int v = t % IMGSZ;      int t1 = t / IMGSZ;
    int u = t1 % IMGSZ;     t1 /= IMGSZ;
    int ch = t1 % 3;        t1 /= 3;
    int prop = t1 % NPROP;  int b = t1 / NPROP;

    int w = (prop < 2) ? 64 : (prop < 5 ? 96 : 128);
    int gi = sel[b * NPROP + prop];
    int x0 = coord[gi * 4 + 0];
    int y0 = coord[gi * 4 + 1];

    float scale = (float)w / (float)IMGSZ;
    float sx = ((float)u + 0.5f) * scale - 0.5f;
    float sy = ((float)v + 0.5f) * scale - 0.5f;
    float fx = floorf(sx), fy = floorf(sy);
    float ax = sx - fx, ay = sy - fy;
    int ix0 = (int)fx, iy0 = (int)fy;
    int ix1 = min(max(ix0 + 1, 0), w - 1);
    int iy1 = min(max(iy0 + 1, 0), w - 1);
    ix0 = min(max(ix0, 0), w - 1);
    iy0 = min(max(iy0, 0), w - 1);

    const float* img = x + ((size_t)b * 3 + ch) * IMGSZ * IMGSZ;
    const float* r0 = img + (size_t)(x0 + ix0) * IMGSZ + y0;
    const float* r1 = img + (size_t)(x0 + ix1) * IMGSZ + y0;
    float p00 = r0[iy0], p01 = r0[iy1];
    float p10 = r1[iy0], p11 = r1[iy1];
    out[t] = (1.f - ax) * ((1.f - ay) * p00 + ay * p01)
           +         ax * ((1.f - ay) * p10 + ay * p11);
}

// ---------------------------------------------------------------- logits GEMM via f32 WMMA
// D(16x16) += A(16x4) * B(4x16); M = batch rows 0..7 (rows 8..15 replicate, unused),
// N tiles of 16 over 200 (cols >=200 clamp to 199, unused), K = 2048 in steps of 4.
// All loads unconditional -> no EXEC masking around the WMMA chain.
__global__ void k_logits(const float* __restrict__ emb, const float* __restrict__ W,
                         const float* __restrict__ bias, float* __restrict__ out) {
    int tile = blockIdx.x;        // 0..12 -> N tile
    int lane = threadIdx.x;       // 0..31 (wave32)
    int half = lane >> 4;         // K half: lanes 0-15 hold K=0,1; lanes 16-31 hold K=2,3
    int lm = lane & 15;           // M index for A, N index for B
    int ncol = tile * 16 + lm;

    int arow = lm & 7;                          // pad rows -> duplicate valid rows (results unused)
    int bcol = (ncol < 200) ? ncol : 199;       // pad cols -> clamp (results unused)

    const float* pa = emb + arow * 2048 + half * 2;             // 8B-aligned (even offset)
    const float* pb = W + (size_t)(half * 2) * 200 + bcol;

    v8f c = {};
    #pragma unroll 4
    for (int k = 0; k < 2048; k += 4) {
        v2f a = *(const v2f*)(pa + k);          // global_load_b64: A[k..k+1] for this lane half
        v2f bm;
        bm[0] = pb[(size_t)k * 200];            // B[k+half*2    ][bcol]
        bm[1] = pb[(size_t)k * 200 + 200];      // B[k+half*2 + 1][bcol]
        // 8 args: (neg_a, A, neg_b, B, c_mod, C, reuse_a, reuse_b)
        c = __builtin_amdgcn_wmma_f32_16x16x4_f32(false, a, false, bm, (short)0, c, false, false);
    }
    // C/D layout: lanes 0-15 -> VGPR v holds M=v, N=lane. Only M=0..7 are real rows.
    if (half == 0 && ncol < 200) {
        float bb = bias[ncol];
        #pragma unroll
        for (int m = 0; m < 8; ++m)
            out[m * 200 + ncol] = c[m] + bb;
    }
}

// ---------------------------------------------------------------- launcher
extern "C" void kernel_launch(void* const* d_in, const int* in_sizes, int n_in,
                              void* d_out, int out_size, void* d_ws, size_t ws_size,
                              hipStream_t stream) {
    const float* x     = (const float*)d_in[0];   // (8,3,448,448)
    const float* fm    = (const float*)d_in[1];   // (8,384,28,28)
    const float* emb   = (const float*)d_in[2];   // (8,2048)
    const float* Wc    = (const float*)d_in[3];   // (2048,200)
    const float* bc    = (const float*)d_in[4];   // (200,)
    const int*   coord = (const int*)d_in[5];     // (1595,4)

    float* out = (float*)d_out;
    float* out_idx = out;                                   // 56
    float* out_sc  = out + 56;                              // 56
    float* out_all = out + 112;                             // 8*1595
    float* out_img = out_all + BN * TOT_WIN;                // 8*7*3*448*448
    float* out_log = out_img + (size_t)BN * NPROP * 3 * IMGSZ * IMGSZ;  // 8*200

    float* s   = (float*)d_ws;                              // 8*28*28 floats
    int*   sel = (int*)((char*)d_ws + BN * FM * FM * sizeof(float));    // 56 ints

    k_sumch <<<(BN * FM * FM + 255) / 256, 256, 0, stream>>>(fm, s);
    k_scores<<<(BN * TOT_WIN + 255) / 256, 256, 0, stream>>>(s, out_all);
    k_nms   <<<BN * 3, 256, 0, stream>>>(out_all, out_idx, out_sc, sel);

    int total = BN * NPROP * 3 * IMGSZ * IMGSZ;
    k_crop  <<<(total + 255) / 256, 256, 0, stream>>>(x, coord, sel, out_img);
    k_logits<<<13, 32, 0, stream>>>(emb, Wc, bc, out_log);
}